// GraphAttention_4020089389722
// MI455X (gfx1250) — compile-verified
//
#include <hip/hip_runtime.h>
#include <stdint.h>

#define N_NODES 4096
#define DIN 512
#define FDIM 64
#define NHEADS 8
#define MASK_WORDS (N_NODES / 32)  // 128 u32 words per adjacency row

#define AS1 __attribute__((address_space(1)))
#define AS3 __attribute__((address_space(3)))

typedef __attribute__((ext_vector_type(2))) float v2f;
typedef __attribute__((ext_vector_type(8))) float v8f;
typedef int v4i __attribute__((vector_size(16)));  // matches builtin param type

// D = A(16x4 f32) * B(4x16 f32) + C(16x16 f32), full f32 precision.
__device__ __forceinline__ v8f wmma_f32(v2f a, v2f b, v8f c) {
  return __builtin_amdgcn_wmma_f32_16x16x4_f32(
      /*neg_a=*/false, a, /*neg_b=*/false, b,
      /*c_mod=*/(short)0, c, /*reuse_a=*/false, /*reuse_b=*/false);
}

// -------------------------------------------------------------------------
// Kernel 0: compress (adj > 0) into a bitmask. One coalesced HBM pass over
// the 64MB adj; output is 2MB that stays hot in L2 for all 8 heads
// (vs. 512MB of per-head f32 re-reads). wave32 ballot builds one u32/wave.
// -------------------------------------------------------------------------
__global__ __launch_bounds__(256) void gat_mask_kernel(
    const float* __restrict__ adj, uint32_t* __restrict__ maskbits) {
  const size_t flat = (size_t)blockIdx.x * 256 + threadIdx.x;
  const int lane = threadIdx.x & 31;
  const bool pred = adj[flat] > 0.0f;
#if __has_builtin(__builtin_amdgcn_ballot_w32)
  const uint32_t bits = __builtin_amdgcn_ballot_w32(pred);
#else
  const uint32_t bits = (uint32_t)__ballot(pred);
#endif
  if (lane == 0) maskbits[flat >> 5] = bits;
}

// -------------------------------------------------------------------------
// Kernel 1: feats[h] = x @ kernels[h]   ([4096,512] x [512,64] per head)
// 8 waves/block; wave w owns rows [i0, i0+16) and all 64 output columns.
// A-frag (16x4 f32): lanes 0-15 hold (K=0,K=1) of row M=lane; lanes 16-31
// hold (K=2,K=3).  C/D layout: lane L holds rows v+8*(L>>4), col L&15.
// -------------------------------------------------------------------------
__global__ __launch_bounds__(256) void gat_feats_kernel(
    const float* __restrict__ x, const float* __restrict__ kernels,
    float* __restrict__ feats) {
  const int h = blockIdx.y;
  const int wave = threadIdx.x >> 5;
  const int lane = threadIdx.x & 31;
  const int m = lane & 15;
  const int khalf = lane >> 4;
  const int i0 = blockIdx.x * 128 + wave * 16;

  const float* __restrict__ xrow = x + (size_t)(i0 + m) * DIN;
  const float* __restrict__ kern = kernels + (size_t)h * DIN * FDIM;

  v8f c0 = {}, c1 = {}, c2 = {}, c3 = {};
  for (int k = 0; k < DIN; k += 4) {
    const int ka = k + 2 * khalf;
    v2f a;
    a.x = xrow[ka];
    a.y = xrow[ka + 1];
    const float* kr0 = kern + (size_t)ka * FDIM + m;  // row ka,   col m
    const float* kr1 = kr0 + FDIM;                    // row ka+1, col m
    v2f b;
    b.x = kr0[0];  b.y = kr1[0];  c0 = wmma_f32(a, b, c0);
    b.x = kr0[16]; b.y = kr1[16]; c1 = wmma_f32(a, b, c1);
    b.x = kr0[32]; b.y = kr1[32]; c2 = wmma_f32(a, b, c2);
    b.x = kr0[48]; b.y = kr1[48]; c3 = wmma_f32(a, b, c3);
  }

  float* fo = feats + (size_t)h * N_NODES * FDIM +
              (size_t)(i0 + 8 * khalf) * FDIM + m;
#pragma unroll
  for (int v = 0; v < 8; ++v) {
    float* p = fo + (size_t)v * FDIM;
    p[0]  = c0[v];
    p[16] = c1[v];
    p[32] = c2[v];
    p[48] = c3[v];
  }
}

// -------------------------------------------------------------------------
// Kernel 2: a_s[h,i] = feats[h,i,:].attn_self[h];  a_n likewise. Trivial.
// -------------------------------------------------------------------------
__global__ __launch_bounds__(256) void gat_scores_kernel(
    const float* __restrict__ feats, const float* __restrict__ attn_self,
    const float* __restrict__ attn_neigh, float* __restrict__ a_s,
    float* __restrict__ a_n) {
  const int idx = blockIdx.x * 256 + threadIdx.x;  // idx = h*N + i
  const int h = idx / N_NODES;
  const float* __restrict__ row = feats + (size_t)idx * FDIM;
  const float* __restrict__ wsv = attn_self + h * FDIM;
  const float* __restrict__ wnv = attn_neigh + h * FDIM;
  float s = 0.0f, t = 0.0f;
#pragma unroll 8
  for (int f = 0; f < FDIM; ++f) {
    const float v = row[f];
    s += v * wsv[f];
    t += v * wnv[f];
  }
  a_s[idx] = s;
  a_n[idx] = t;
}

// -------------------------------------------------------------------------
// Kernel 3: fused masked-Gaussian attention aggregate.
// Block = 8 waves = 128 output rows of one head. Each feats j-tile (16x64)
// is staged into LDS ONCE per block (async global->LDS) and consumed by all
// 8 waves -> 8x less L2 traffic than per-wave fetching.
// Each wave: builds W(16x16) in registers (exp * maskbit, accumulating
// row-sums), runs 4 K-slices x 4 N-tiles of f32 WMMA per j-tile.
// Row-normalize (1/(sum+eps)) + bias + ReLU applied at the end.
// LDS rows padded to 72 floats: rows r and r+2 (the two half-wave groups of
// one ds_load) map to disjoint bank sets.
// -------------------------------------------------------------------------
#define AGG_WAVES 8
#define LDS_STRIDE 72

__global__ __launch_bounds__(32 * AGG_WAVES) void gat_aggregate_kernel(
    const uint32_t* __restrict__ maskbits, const float* __restrict__ feats,
    const float* __restrict__ a_s, const float* __restrict__ a_n,
    const float* __restrict__ biases, float* __restrict__ out) {
  __shared__ __align__(16) float tile[16 * LDS_STRIDE];
  __shared__ float rs[AGG_WAVES * 32];

  const int h = blockIdx.y;
  const int wave = threadIdx.x >> 5;
  const int lane = threadIdx.x & 31;
  const int m = lane & 15;
  const int khalf = lane >> 4;
  const int i0 = blockIdx.x * (AGG_WAVES * 16) + wave * 16;
  const int irow = i0 + m;

  const float as_m = a_s[h * N_NODES + irow];
  const uint32_t* __restrict__ mrow = maskbits + (size_t)irow * MASK_WORDS;
  const float* __restrict__ an = a_n + h * N_NODES;
  const float* __restrict__ fh = feats + (size_t)h * N_NODES * FDIM;

  // Staging slot for this thread: row sr (0..15), 4 floats at col sc.
  const int sr = threadIdx.x >> 4;
  const int sc = (threadIdx.x & 15) * 4;

  v8f c0 = {}, c1 = {}, c2 = {}, c3 = {};
  float rsum = 0.0f;

  for (int j0 = 0; j0 < N_NODES; j0 += 16) {
    __syncthreads();  // previous tile fully consumed by all waves
    const float* gsrc = fh + (size_t)(j0 + sr) * FDIM + sc;
    float* ldst = &tile[sr * LDS_STRIDE + sc];
#if __has_builtin(__builtin_amdgcn_global_load_async_to_lds_b128)
    __builtin_amdgcn_global_load_async_to_lds_b128(
        (AS1 v4i*)gsrc, (AS3 v4i*)ldst, /*offset=*/0, /*cpol=*/0);
#if __has_builtin(__builtin_amdgcn_s_wait_asynccnt)
    __builtin_amdgcn_s_wait_asynccnt(0);
#else
    asm volatile("s_wait_asynccnt 0x0" ::: "memory");
#endif
#else
    const float4 v4 = *(const float4*)gsrc;
    *(float4*)ldst = v4;
#endif
    __syncthreads();  // tile visible to all waves

    const uint32_t mword = mrow[j0 >> 5];
    const uint32_t tbits = mword >> (j0 & 31);  // low 16 bits = this tile
#pragma unroll
    for (int kk = 0; kk < 4; ++kk) {
      const int cbase = 4 * kk + 2 * khalf;  // this lane's two K columns
      const int jc = j0 + cbase;
      const float d0 = as_m - an[jc];
      const float d1 = as_m - an[jc + 1];
      const float m0 = (float)((tbits >> cbase) & 1u);
      const float m1 = (float)((tbits >> (cbase + 1)) & 1u);
      const float w0 = __expf(-d0 * d0) * m0;
      const float w1 = __expf(-d1 * d1) * m1;
      rsum += w0 + w1;
      v2f a;
      a.x = w0;
      a.y = w1;
      const float* fr0 = &tile[cbase * LDS_STRIDE + m];
      const float* fr1 = fr0 + LDS_STRIDE;
      v2f b;
      b.x = fr0[0];  b.y = fr1[0];  c0 = wmma_f32(a, b, c0);
      b.x = fr0[16]; b.y = fr1[16]; c1 = wmma_f32(a, b, c1);
      b.x = fr0[32]; b.y = fr1[32]; c2 = wmma_f32(a, b, c2);
      b.x = fr0[48]; b.y = fr1[48]; c3 = wmma_f32(a, b, c3);
    }
  }

  // Row-sum for row r lives split across lanes r and r+16 of this wave.
  float* wrs = rs + wave * 32;
  wrs[lane] = rsum;
  __syncthreads();

  const float eps = 2.2204460492503131e-16f;  // np.finfo(float64).eps
  float inv[8];
#pragma unroll
  for (int v = 0; v < 8; ++v) {
    const int r = 8 * khalf + v;  // row index held in D VGPR v
    inv[v] = 1.0f / (wrs[r] + wrs[r + 16] + eps);
  }

  const float bb0 = biases[h * FDIM + m];
  const float bb1 = biases[h * FDIM + m + 16];
  const float bb2 = biases[h * FDIM + m + 32];
  const float bb3 = biases[h * FDIM + m + 48];

  float* orow = out + (size_t)(i0 + 8 * khalf) * (NHEADS * FDIM) + h * FDIM + m;
#pragma unroll
  for (int v = 0; v < 8; ++v) {
    float* p = orow + (size_t)v * (NHEADS * FDIM);
    p[0]  = fmaxf(c0[v] * inv[v] + bb0, 0.0f);
    p[16] = fmaxf(c1[v] * inv[v] + bb1, 0.0f);
    p[32] = fmaxf(c2[v] * inv[v] + bb2, 0.0f);
    p[48] = fmaxf(c3[v] * inv[v] + bb3, 0.0f);
  }
}

extern "C" void kernel_launch(void* const* d_in, const int* in_sizes, int n_in,
                              void* d_out, int out_size, void* d_ws,
                              size_t ws_size, hipStream_t stream) {
  const float* x          = (const float*)d_in[0];  // [N, DIN]
  const float* adj        = (const float*)d_in[1];  // [N, N]
  const float* kernels    = (const float*)d_in[2];  // [H, DIN, F]
  const float* attn_self  = (const float*)d_in[3];  // [H, F]
  const float* attn_neigh = (const float*)d_in[4];  // [H, F]
  const float* biases     = (const float*)d_in[5];  // [H, F]
  float* out = (float*)d_out;                       // [N, H*F]

  float* ws = (float*)d_ws;
  float* feats = ws;                                     // H*N*F floats (8 MB)
  float* a_s = feats + (size_t)NHEADS * N_NODES * FDIM;  // H*N floats
  float* a_n = a_s + (size_t)NHEADS * N_NODES;           // H*N floats
  uint32_t* maskbits = (uint32_t*)(a_n + (size_t)NHEADS * N_NODES);  // 2 MB

  (void)in_sizes; (void)n_in; (void)out_size; (void)ws_size;

  gat_mask_kernel<<<(N_NODES * (size_t)N_NODES) / 256, 256, 0, stream>>>(
      adj, maskbits);
  gat_feats_kernel<<<dim3(N_NODES / 128, NHEADS), 256, 0, stream>>>(
      x, kernels, feats);
  gat_scores_kernel<<<(NHEADS * N_NODES) / 256, 256, 0, stream>>>(
      feats, attn_self, attn_neigh, a_s, a_n);
  gat_aggregate_kernel<<<dim3(N_NODES / 128, NHEADS), 32 * AGG_WAVES, 0,
                         stream>>>(maskbits, feats, a_s, a_n, biases, out);
}